// DetectionLoss_77129022701614
// MI455X (gfx1250) — compile-verified
//
#include <hip/hip_runtime.h>
#include <stdint.h>

#define A_N 65536
#define T_N 32
#define B_N 16
#define EPSF 1e-6f

typedef float    v2f __attribute__((ext_vector_type(2)));
typedef float    v8f __attribute__((ext_vector_type(8)));
typedef uint32_t v4u __attribute__((ext_vector_type(4)));
typedef int      v4i __attribute__((ext_vector_type(4)));
typedef int      v8i __attribute__((ext_vector_type(8)));

#if defined(__has_builtin)
#if __has_builtin(__builtin_amdgcn_tensor_load_to_lds)
#define HAVE_TDM 1
#endif
#endif

// Compile-time probe: report (in stderr) which K3 LDS-fill path the DEVICE
// pass selected, so the compile-only loop can verify tensor_load_to_lds usage.
#if defined(__HIP_DEVICE_COMPILE__)
#if defined(HAVE_TDM)
#if __clang_major__ >= 23
#warning "CDNA5 probe: device pass uses 6-arg __builtin_amdgcn_tensor_load_to_lds"
#else
#warning "CDNA5 probe: device pass uses 5-arg __builtin_amdgcn_tensor_load_to_lds"
#endif
#else
#warning "CDNA5 probe: device pass has NO tensor_load_to_lds builtin - fallback copy"
#endif
#endif

// Wave-level sum via V_WMMA_F32_16X16X4_F32 with A = ones(16x4).
// D[m][n] = sum_k B[k][n]; a full 32-lane reduce of D-VGPR0 covers rows M=0 and
// M=8, i.e. 2 * (sum of all B entries), independent of exact B lane layout
// (every B element lands in exactly one column sum). Requires full-wave EXEC.
__device__ __forceinline__ float wave_sum_wmma(float v) {
  v2f a; a[0] = 1.0f; a[1] = 1.0f;
  v2f b; b[0] = v;    b[1] = 0.0f;
  v8f c = {};
  c = __builtin_amdgcn_wmma_f32_16x16x4_f32(false, a, false, b, (short)0, c, false, false);
  float t = c[0];
  t += __shfl_xor(t, 16, 32);
  t += __shfl_xor(t, 8, 32);
  t += __shfl_xor(t, 4, 32);
  t += __shfl_xor(t, 2, 32);
  t += __shfl_xor(t, 1, 32);
  return t * 0.5f;
}

__device__ __forceinline__ float block_sum(float v, float* sw, int tid, int nwaves) {
  float w = wave_sum_wmma(v);
  int wave = tid >> 5;
  int lane = tid & 31;
  if (lane == 0) sw[wave] = w;
  __syncthreads();
  if (wave == 0) {                       // wave-uniform branch, full EXEC in wave 0
    float x = (lane < nwaves) ? sw[lane] : 0.0f;
    x = wave_sum_wmma(x);
    if (lane == 0) sw[0] = x;
  }
  __syncthreads();
  float r = sw[0];
  __syncthreads();
  return r;
}

// ---------------------------------------------------------------- K0: zero ws
__global__ void k0_zero(uint32_t* p, int n) {
  int i = blockIdx.x * blockDim.x + threadIdx.x;
  if (i < n) p[i] = 0u;
}

// ------------------------------------------- K1: per-target top-3 + forced set
__global__ void k1_topk(const float* __restrict__ anchors,
                        const float* __restrict__ tboxes,
                        uint32_t* __restrict__ forced) {
  __shared__ float sv[256 * 3];
  __shared__ int   si[256 * 3];
  __shared__ int   scnt;
  int tid = threadIdx.x;
  int b = blockIdx.x >> 5;   // / T_N
  int t = blockIdx.x & 31;   // % T_N
  if (tid == 0) scnt = 0;
  __syncthreads();
  const float* tb = tboxes + ((size_t)b * T_N + t) * 4;
  float tx1 = tb[0], ty1 = tb[1], tx2 = tb[2], ty2 = tb[3];
  float areaT = (tx2 - tx1) * (ty2 - ty1);
  float v0 = -1.f, v1 = -1.f, v2 = -1.f;
  int   i0 = -1, i1 = -1, i2 = -1;
  int cnt = 0;
  for (int a = tid; a < A_N; a += 256) {
    // gfx1250 global_prefetch_b8 for the next strided tile of the anchor array
    if (a + 256 < A_N) __builtin_prefetch(&((const float4*)anchors)[a + 256], 0, 1);
    float4 ab = ((const float4*)anchors)[a];
    float areaA = (ab.z - ab.x) * (ab.w - ab.y);
    float ix1 = fmaxf(ab.x, tx1), iy1 = fmaxf(ab.y, ty1);
    float ix2 = fminf(ab.z, tx2), iy2 = fminf(ab.w, ty2);
    float inter = fmaxf(ix2 - ix1, 0.f) * fmaxf(iy2 - iy1, 0.f);
    float iou = inter / (areaA + areaT - inter + EPSF);
    cnt += (iou > 0.3f) ? 1 : 0;
    if (iou > v0)      { v2 = v1; i2 = i1; v1 = v0; i1 = i0; v0 = iou; i0 = a; }
    else if (iou > v1) { v2 = v1; i2 = i1; v1 = iou; i1 = a; }
    else if (iou > v2) { v2 = iou; i2 = a; }
  }
  sv[tid * 3 + 0] = v0; si[tid * 3 + 0] = i0;
  sv[tid * 3 + 1] = v1; si[tid * 3 + 1] = i1;
  sv[tid * 3 + 2] = v2; si[tid * 3 + 2] = i2;
  atomicAdd(&scnt, cnt);
  __syncthreads();
  if (tid == 0) {
    float tv0 = -2.f, tv1 = -2.f, tv2 = -2.f;
    int   ti0 = -1, ti1 = -1, ti2 = -1;
    for (int e = 0; e < 768; ++e) {
      float v = sv[e]; int i = si[e];
      if (i < 0) continue;
      if (v > tv0 || (v == tv0 && (unsigned)i < (unsigned)ti0)) {
        tv2 = tv1; ti2 = ti1; tv1 = tv0; ti1 = ti0; tv0 = v; ti0 = i;
      } else if (v > tv1 || (v == tv1 && (unsigned)i < (unsigned)ti1)) {
        tv2 = tv1; ti2 = ti1; tv1 = v; ti1 = i;
      } else if (v > tv2 || (v == tv2 && (unsigned)i < (unsigned)ti2)) {
        tv2 = v; ti2 = i;
      }
    }
    uint32_t* fb = forced + (size_t)b * (A_N / 32);
    if (tv0 > 0.3f && ti0 >= 0) atomicOr(&fb[ti0 >> 5], 1u << (ti0 & 31));
    if (tv1 > 0.3f && ti1 >= 0) atomicOr(&fb[ti1 >> 5], 1u << (ti1 & 31));
    if (tv2 > 0.3f && ti2 >= 0) atomicOr(&fb[ti2 >> 5], 1u << (ti2 & 31));
    if (scnt == 0 && ti0 >= 0)  atomicOr(&fb[ti0 >> 5], 1u << (ti0 & 31));
  }
}

// ---------------- K2: per-anchor fused IoU max / focal / GIoU / accumulation
__global__ void k2_anchor(const float* __restrict__ bbox_pred,
                          const float* __restrict__ conf_pred,
                          const float* __restrict__ anchors,
                          const float* __restrict__ tboxes,
                          const uint32_t* __restrict__ forced,
                          float* __restrict__ flneg,
                          float* __restrict__ scal) {
  __shared__ float tb[T_N * 4];
  int tid = threadIdx.x;
  int b = blockIdx.x >> 8;                 // 256 blocks per image
  int a = ((blockIdx.x & 255) << 8) + tid;
  if (tid < T_N * 4) tb[tid] = tboxes[(size_t)b * T_N * 4 + tid];
  __syncthreads();

  float4 ab = ((const float4*)anchors)[a];
  float areaA = (ab.z - ab.x) * (ab.w - ab.y);
  float best = -1.0f; int bt = 0;
  #pragma unroll 4
  for (int t = 0; t < T_N; ++t) {
    float tx1 = tb[4 * t], ty1 = tb[4 * t + 1], tx2 = tb[4 * t + 2], ty2 = tb[4 * t + 3];
    float ix1 = fmaxf(ab.x, tx1), iy1 = fmaxf(ab.y, ty1);
    float ix2 = fminf(ab.z, tx2), iy2 = fminf(ab.w, ty2);
    float inter = fmaxf(ix2 - ix1, 0.f) * fmaxf(iy2 - iy1, 0.f);
    float areaT = (tx2 - tx1) * (ty2 - ty1);
    float iou = inter / (areaA + areaT - inter + EPSF);
    if (iou > best) { best = iou; bt = t; }
  }
  bool isF = (forced[(size_t)b * (A_N / 32) + (a >> 5)] >> (a & 31)) & 1u;
  bool pos = (best >= 0.5f) || isF;
  bool neg = (best < 0.4f) && !pos;

  float p  = conf_pred[(size_t)b * A_N + a];
  float pt = pos ? p : 1.0f - p;
  float af = pos ? 0.25f : 0.75f;
  float ptc = fminf(fmaxf(pt, 1e-6f), 1.0f - 1e-6f);
  float om = 1.0f - pt;
  float fl = -af * om * om * logf(ptc);
  flneg[(size_t)b * A_N + a] = neg ? fl : 0.0f;   // bits==0 for non-negatives

  float4 bp = ((const float4*)bbox_pred)[(size_t)b * A_N + a];
  float mx1 = tb[4 * bt], my1 = tb[4 * bt + 1], mx2 = tb[4 * bt + 2], my2 = tb[4 * bt + 3];
  float ix1 = fmaxf(bp.x, mx1), iy1 = fmaxf(bp.y, my1);
  float ix2 = fminf(bp.z, mx2), iy2 = fminf(bp.w, my2);
  float inter = fmaxf(ix2 - ix1, 0.f) * fmaxf(iy2 - iy1, 0.f);
  float a1 = (bp.z - bp.x) * (bp.w - bp.y);
  float a2 = (mx2 - mx1) * (my2 - my1);
  float uni = a1 + a2 - inter;
  float iou = inter / (uni + EPSF);
  float ex1 = fminf(bp.x, mx1), ey1 = fminf(bp.y, my1);
  float ex2 = fmaxf(bp.z, mx2), ey2 = fmaxf(bp.w, my2);
  float enc = (ex2 - ex1) * (ey2 - ey1);
  float giou = iou - (enc - uni) / (enc + EPSF);
  float l1 = (fabsf(bp.x - mx1) + fabsf(bp.y - my1) +
              fabsf(bp.z - mx2) + fabsf(bp.w - my2)) * 0.25f;
  float per = (1.0f - giou) + 0.5f * l1;

  float s_np = wave_sum_wmma(pos ? 1.0f : 0.0f);
  float s_nn = wave_sum_wmma(neg ? 1.0f : 0.0f);
  float s_fl = wave_sum_wmma(pos ? fl  : 0.0f);
  float s_bx = wave_sum_wmma(pos ? per : 0.0f);
  if ((tid & 31) == 0) {
    atomicAdd(&scal[0 * 16 + b], s_np);
    atomicAdd(&scal[1 * 16 + b], s_nn);
    atomicAdd(&scal[2 * 16 + b], s_fl);
    atomicAdd(&scal[3 * 16 + b], s_bx);
  }
}

// ------ K3: per-image exact hard-negative selection in 256KB LDS (TDM fill)
__global__ void __launch_bounds__(1024)
k3_select(const float* __restrict__ flneg, float* __restrict__ scal) {
  extern __shared__ uint32_t smem[];
  uint32_t* flb = smem;                    // 65536 u32 = 256 KB
  float*    sw  = (float*)(smem + A_N);    // 32-float reduction scratch
  int tid = threadIdx.x;
  int b = blockIdx.x;
  const uint32_t* src = (const uint32_t*)(flneg + (size_t)b * A_N);

#if defined(HAVE_TDM)
  if (tid < 32) {                          // one wave issues the DMA
    uint64_t ga = (uint64_t)(uintptr_t)src;
    uint32_t lds = (uint32_t)(uintptr_t)flb;
    v4u g0;
    g0[0] = 1u;                                            // count=1 valid descriptor
    g0[1] = lds;                                           // lds_addr
    g0[2] = (uint32_t)(ga & 0xFFFFFFFFu);                  // global_addr[31:0]
    g0[3] = ((uint32_t)(ga >> 32) & 0x01FFFFFFu) | 0x80000000u; // addr[56:32] | type=2
    v8i g1;
    g1[0] = 0x00020000;   // workgroup_mask=0, data_size=2 (4B)
    g1[1] = 0x40000000;   // tensor_dim0 = 16384 (low16 << 16)
    g1[2] = 0x00040000;   // tensor_dim0 hi=0, tensor_dim1 = 4 (low16 << 16)
    g1[3] = 0x40000000;   // tensor_dim1 hi=0, tile_dim0 = 16384 (<< 16)
    g1[4] = 4;            // tile_dim1 = 4, tile_dim2 = 0
    g1[5] = 16384;        // tensor_dim0_stride low32
    g1[6] = 0;
    g1[7] = 0;
    v4i gz = {0, 0, 0, 0};
#if __clang_major__ >= 23
    v8i gz8 = {0, 0, 0, 0, 0, 0, 0, 0};
    __builtin_amdgcn_tensor_load_to_lds(g0, g1, gz, gz, gz8, 0);
#else
    __builtin_amdgcn_tensor_load_to_lds(g0, g1, gz, gz, 0);
#endif
    __builtin_amdgcn_s_wait_tensorcnt(0);
  }
#else
  for (int j = 0; j < 64; ++j) flb[tid + (j << 10)] = src[tid + (j << 10)];
#endif
  __syncthreads();

  float np_f = scal[b], nn_f = scal[16 + b], sfp = scal[32 + b], sbp = scal[48 + b];
  uint32_t np = (uint32_t)(np_f + 0.5f);
  uint32_t nn = (uint32_t)(nn_f + 0.5f);
  uint32_t k;
  if (np > 0) {
    uint32_t quo = (uint32_t)A_N / np;
    uint32_t maxneg = (quo < 3u ? quo : 3u) * np;
    k = nn < maxneg ? nn : maxneg;
  } else {
    k = nn < 100u ? nn : 100u;             // min(A/4, 100) == 100
  }

  float sum_sel = 0.0f;
  if (k > 0) {
    // Binary search over float bit patterns (all values >= 0): largest thr
    // with count(bits >= thr) >= k  ==  k-th largest focal loss.
    uint32_t lo = 0u, hi = 0x7F7FFFFFu;
    while (lo < hi) {
      uint32_t mid = lo + ((hi - lo + 1u) >> 1);
      int c = 0;
      for (int j = 0; j < 64; ++j) c += (flb[tid + (j << 10)] >= mid) ? 1 : 0;
      float tot = block_sum((float)c, sw, tid, 32);
      uint32_t cnt = (uint32_t)(tot + 0.5f);
      if (cnt >= k) lo = mid; else hi = mid - 1u;
    }
    uint32_t thr = lo;
    float s = 0.0f; int c = 0;
    for (int j = 0; j < 64; ++j) {
      uint32_t u = flb[tid + (j << 10)];
      if (u > thr) { s += __uint_as_float(u); c++; }
    }
    float sum_gt = block_sum(s, sw, tid, 32);
    float cnt_gt = block_sum((float)c, sw, tid, 32);
    sum_sel = sum_gt + ((float)k - cnt_gt) * __uint_as_float(thr);  // exact w/ ties
  }
  if (tid == 0) {
    uint32_t lcnt = np + k;
    float conf = (sfp + sum_sel) / (float)(lcnt > 0 ? lcnt : 1u);
    float bbox = (np > 0) ? sbp / (float)np : 0.0f;
    scal[64 + b] = conf;
    scal[80 + b] = bbox;
  }
}

// ---------------------------------------------------------------- K4: finalize
__global__ void k4_final(const float* __restrict__ scal, float* __restrict__ out) {
  if (threadIdx.x == 0) {
    float cs = 0.f, bs = 0.f;
    for (int b = 0; b < B_N; ++b) { cs += scal[64 + b]; bs += scal[80 + b]; }
    cs *= (1.0f / 16.0f);
    bs *= (1.0f / 16.0f);
    out[0] = cs + bs;
    out[1] = cs;
    out[2] = bs;
  }
}

extern "C" void kernel_launch(void* const* d_in, const int* in_sizes, int n_in,
                              void* d_out, int out_size, void* d_ws, size_t ws_size,
                              hipStream_t stream) {
  const float* bbox_pred = (const float*)d_in[0];   // [B,A,4]
  const float* conf_pred = (const float*)d_in[1];   // [B,A]
  const float* anchors   = (const float*)d_in[2];   // [A,4]
  const float* tboxes    = (const float*)d_in[3];   // [B,T,4]
  uint8_t* ws = (uint8_t*)d_ws;
  uint32_t* forced = (uint32_t*)ws;                 // 16 * 2048 u32 = 128 KB
  float*    scal   = (float*)(ws + 131072);         // 128 floats (6 x 16)
  float*    flneg  = (float*)(ws + 131072 + 512);   // 16 * 65536 f32 = 4 MB
  float*    out    = (float*)d_out;

  k0_zero<<<(32768 + 128 + 255) / 256, 256, 0, stream>>>((uint32_t*)ws, 32768 + 128);
  k1_topk<<<B_N * T_N, 256, 0, stream>>>(anchors, tboxes, forced);
  k2_anchor<<<B_N * (A_N / 256), 256, 0, stream>>>(bbox_pred, conf_pred, anchors,
                                                   tboxes, forced, flneg, scal);
  k3_select<<<B_N, 1024, A_N * 4 + 128, stream>>>(flneg, scal);
  k4_final<<<1, 32, 0, stream>>>(scal, out);
}